// Set2SetPooling_33861522162214
// MI455X (gfx1250) — compile-verified
//
#include <hip/hip_runtime.h>
#include <math.h>

#define DIM 128
#define NSEG 2048
#define NSTEPS 4
#define KPACK 384   // q_star | r | h
#define NGATE 512   // 4*DIM
#define CH 64       // rows per async-staged chunk (32KB per buffer)

typedef __attribute__((ext_vector_type(2))) float v2f;
typedef __attribute__((ext_vector_type(8))) float v8f;

// ---------------------------------------------------------------------------
// K0: segment pointers from sorted batch (int64). seg_ptr[b] = first n with
// batch[n] >= b; seg_ptr[NSEG] = N. Every entry written exactly once.
// ---------------------------------------------------------------------------
__global__ void build_seg_ptr(const long long* __restrict__ batch,
                              int* __restrict__ seg_ptr, int N) {
  int n = blockIdx.x * blockDim.x + threadIdx.x;
  if (n >= N) return;
  int b = (int)batch[n];
  int bprev = (n == 0) ? -1 : (int)batch[n - 1];
  for (int bb = bprev + 1; bb <= b; ++bb) seg_ptr[bb] = n;
  if (n == N - 1)
    for (int bb = b + 1; bb <= NSEG; ++bb) seg_ptr[bb] = N;
}

// ---------------------------------------------------------------------------
// K1: q_star = segment mean of x; zero h, c. One block per segment, one
// thread per dim (coalesced 512B rows).
// ---------------------------------------------------------------------------
__global__ void seg_mean(const float* __restrict__ x, const int* __restrict__ seg_ptr,
                         float* __restrict__ q_star, float* __restrict__ h,
                         float* __restrict__ c) {
  int b = blockIdx.x;
  int d = threadIdx.x;  // 128 threads
  int s = seg_ptr[b], e = seg_ptr[b + 1];
  float acc = 0.f;
  for (int n = s; n < e; ++n) {
    __builtin_prefetch(&x[(size_t)(n + 8) * DIM + d], 0, 0);
    acc += x[(size_t)n * DIM + d];
  }
  float cnt = (float)(e - s);
  q_star[b * DIM + d] = acc / fmaxf(cnt, 1.0f);
  h[b * DIM + d] = 0.f;
  c[b * DIM + d] = 0.f;
}

// ---------------------------------------------------------------------------
// K2: pack [w_ih | w_hh] -> w_pack[512][384]; bias_pack = b_ih + b_hh.
// ---------------------------------------------------------------------------
__global__ void pack_weights(const float* __restrict__ w_ih, const float* __restrict__ w_hh,
                             const float* __restrict__ b_ih, const float* __restrict__ b_hh,
                             float* __restrict__ w_pack, float* __restrict__ bias_pack) {
  int n = blockIdx.x;   // 512 gate rows
  int t = threadIdx.x;  // 128
  for (int k = t; k < 2 * DIM; k += 128) w_pack[n * KPACK + k] = w_ih[n * (2 * DIM) + k];
  for (int k = t; k < DIM; k += 128)     w_pack[n * KPACK + 2 * DIM + k] = w_hh[n * DIM + k];
  if (t == 0) bias_pack[n] = b_ih[n] + b_hh[n];
}

// ---------------------------------------------------------------------------
// fp32 WMMA GEMM:  Out[M,N] = A[M,K] @ Bmat[K,N]
//   BT = true : Bmat[k][n] = Bsrc[k*ldb + n]
//   BT = false: Bmat[k][n] = Bsrc[n*ldb + k]   (W^T GEMM)
// One wave per 16x16 output tile, V_WMMA_F32_16X16X4_F32 over K in steps of 4.
// ---------------------------------------------------------------------------
template <bool BT>
__global__ void wmma_gemm_f32(const float* __restrict__ A, const float* __restrict__ Bsrc,
                              float* __restrict__ Out, int M, int N, int K,
                              int lda, int ldb) {
  const int lane = threadIdx.x & 31;
  const int wave = threadIdx.x >> 5;
  const int half = lane >> 4;
  const int l16  = lane & 15;
  const int tiles_n = N >> 4;
  const int wg = blockIdx.x * (blockDim.x >> 5) + wave;
  const int tm = wg / tiles_n;
  const int tn = wg % tiles_n;
  const int row = tm * 16 + l16;
  const int col = tn * 16 + l16;

  v8f acc = {};
  for (int k0 = 0; k0 < K; k0 += 4) {
    const int ka = k0 + half * 2;
    v2f a;
    a.x = A[(size_t)row * lda + ka];
    a.y = A[(size_t)row * lda + ka + 1];
    v2f bv;
    if (BT) {
      bv.x = Bsrc[(size_t)ka * ldb + col];
      bv.y = Bsrc[(size_t)(ka + 1) * ldb + col];
    } else {
      bv.x = Bsrc[(size_t)col * ldb + ka];
      bv.y = Bsrc[(size_t)col * ldb + ka + 1];
    }
    acc = __builtin_amdgcn_wmma_f32_16x16x4_f32(
        /*neg_a=*/false, a, /*neg_b=*/false, bv,
        /*c_mod=*/(short)0, acc, /*reuse_a=*/false, /*reuse_b=*/false);
  }
#pragma unroll
  for (int j = 0; j < 8; ++j) {
    int r = tm * 16 + j + half * 8;
    Out[(size_t)r * N + col] = acc[j];
  }
}

// ---------------------------------------------------------------------------
// K3: fused attention step for one segment per block (128 threads = 4 waves).
//   scores[n] = x[n].v_b + att_b.q_b ; segment softmax ; r_b = sum w_n * x[n]
// Phase 3 stages x rows into LDS via async global->LDS copies (ASYNCcnt),
// double-buffered: one global_load_async_to_lds_b128 per wave moves one
// 512B row (32 lanes x 16B).
// ---------------------------------------------------------------------------
__global__ void attention_step(const float* __restrict__ x, const int* __restrict__ seg_ptr,
                               const float* __restrict__ v, const float* __restrict__ att_b,
                               const float* __restrict__ q_star,
                               float* __restrict__ scores, float* __restrict__ r) {
  __shared__ float s_v[DIM];
  __shared__ float s_red[4];
  __shared__ float s_w[CH];
  __shared__ __align__(16) float s_x[2 * CH * DIM];  // 64KB double buffer

  const int b = blockIdx.x;
  const int tid = threadIdx.x;
  const int lane = tid & 31;
  const int wave = tid >> 5;
  const int s = seg_ptr[b], e = seg_ptr[b + 1];

  // sbias = att_b . q_star[b]; stage v[b] in LDS
  s_v[tid] = v[b * DIM + tid];
  float psb = att_b[tid] * q_star[b * DIM + tid];
  for (int off = 16; off; off >>= 1) psb += __shfl_xor(psb, off, 32);
  if (lane == 0) s_red[wave] = psb;
  __syncthreads();
  const float sbias = s_red[0] + s_red[1] + s_red[2] + s_red[3];
  __syncthreads();  // s_red reuse below

  if (e <= s) { r[b * DIM + tid] = 0.f; return; }  // empty segment (uniform)

  // ---- phase 1: scores + block max. One node per wave, lane covers 4 dims.
  const float4 vloc = ((const float4*)s_v)[lane];
  float wmax = -3.402823466e38f;
  for (int n = s + wave; n < e; n += 4) {
    __builtin_prefetch(&x[(size_t)(n + 4) * DIM] + lane * 4, 0, 0);
    float4 xr = ((const float4*)(x + (size_t)n * DIM))[lane];
    float d = xr.x * vloc.x + xr.y * vloc.y + xr.z * vloc.z + xr.w * vloc.w;
    for (int off = 16; off; off >>= 1) d += __shfl_xor(d, off, 32);
    float sc = d + sbias;   // identical across lanes after butterfly
    if (lane == 0) scores[n] = sc;
    wmax = fmaxf(wmax, sc);
  }
  if (lane == 0) s_red[wave] = wmax;
  __syncthreads();
  const float m = fmaxf(fmaxf(s_red[0], s_red[1]), fmaxf(s_red[2], s_red[3]));

  // ---- phase 2: denom (scores are L2-hot, 4MB total)
  float psum = 0.f;
  for (int n = s + tid; n < e; n += 128) psum += expf(scores[n] - m);
  for (int off = 16; off; off >>= 1) psum += __shfl_xor(psum, off, 32);
  __syncthreads();
  if (lane == 0) s_red[wave] = psum;
  __syncthreads();
  const float denom = s_red[0] + s_red[1] + s_red[2] + s_red[3];
  const float inv = (denom > 0.f) ? 1.f / denom : 1.f;  // reference guard

  // ---- phase 3: r_b[d] = sum_n w_n * x[n][d]; async-stage rows into LDS.
  const int nchunks = (e - s + CH - 1) / CH;
  const unsigned lbase = (unsigned)(unsigned long long)&s_x[0];

  // issue async copy of chunk ci into buffer (ci&1); wave w copies rows
  // w, w+4, ..., each as a single 512B b128 transfer (lane = 16B).
  auto issue_chunk = [&](int ci) {
    const int c0 = s + ci * CH;
    const int cnt = min(CH, e - c0);
    const unsigned bufb = lbase + (unsigned)((ci & 1) * CH * DIM * 4);
#pragma unroll
    for (int i = 0; i < CH / 4; ++i) {
      const int row = i * 4 + wave;       // uniform per wave
      if (row < cnt) {
        unsigned ldst = bufb + (unsigned)(row * DIM * 4) + (unsigned)(lane * 16);
        const float* g = x + (size_t)(c0 + row) * DIM + lane * 4;
        asm volatile("global_load_async_to_lds_b128 %0, %1, off"
                     :: "v"(ldst), "v"(g) : "memory");
      }
    }
  };

  issue_chunk(0);
  float racc = 0.f;
  for (int ci = 0; ci < nchunks; ++ci) {
    const int c0 = s + ci * CH;
    const int cnt = min(CH, e - c0);
    if (ci + 1 < nchunks) {
      issue_chunk(ci + 1);                              // fills other buffer
      asm volatile("s_wait_asynccnt 0x10" ::: "memory"); // chunk ci complete
    } else {
      asm volatile("s_wait_asynccnt 0x0" ::: "memory");
    }
    if (tid < cnt) s_w[tid] = expf(scores[c0 + tid] - m) * inv;
    __syncthreads();  // chunk ci resident (all waves) + s_w visible
    const float* xc = &s_x[(ci & 1) * CH * DIM];
    for (int j = 0; j < cnt; ++j)
      racc += s_w[j] * xc[j * DIM + tid];
    __syncthreads();  // done reading buffer before it is refilled (ci+2)
  }
  r[b * DIM + tid] = racc;
}

// ---------------------------------------------------------------------------
// K4: pack lstm_in = [q_star | r | h]  (2048 x 384)
// ---------------------------------------------------------------------------
__global__ void pack_lstm_in(const float* __restrict__ q_star, const float* __restrict__ r,
                             const float* __restrict__ h, float* __restrict__ lstm_in) {
  int b = blockIdx.x, t = threadIdx.x;  // 128
  lstm_in[b * KPACK + t]            = q_star[b * DIM + t];
  lstm_in[b * KPACK + DIM + t]      = r[b * DIM + t];
  lstm_in[b * KPACK + 2 * DIM + t]  = h[b * DIM + t];
}

// ---------------------------------------------------------------------------
// K5: LSTM pointwise: gates -> (h, c); q_star = h_new.
// ---------------------------------------------------------------------------
__global__ void lstm_act(const float* __restrict__ gates, const float* __restrict__ bias_pack,
                         float* __restrict__ h, float* __restrict__ c,
                         float* __restrict__ q_star) {
  int idx = blockIdx.x * blockDim.x + threadIdx.x;  // NSEG*DIM
  int b = idx >> 7, d = idx & 127;
  const float* g = gates + (size_t)b * NGATE;
  float gi = g[d]           + bias_pack[d];
  float gf = g[DIM + d]     + bias_pack[DIM + d];
  float gg = g[2 * DIM + d] + bias_pack[2 * DIM + d];
  float go = g[3 * DIM + d] + bias_pack[3 * DIM + d];
  float i_ = 1.f / (1.f + expf(-gi));
  float f_ = 1.f / (1.f + expf(-gf));
  float g_ = tanhf(gg);
  float o_ = 1.f / (1.f + expf(-go));
  float cn = f_ * c[idx] + i_ * g_;
  float hn = o_ * tanhf(cn);
  c[idx] = cn;
  h[idx] = hn;
  q_star[idx] = hn;
}

// ---------------------------------------------------------------------------
// K6: out = [h | r]  (2048 x 256)
// ---------------------------------------------------------------------------
__global__ void write_out(const float* __restrict__ h, const float* __restrict__ r,
                          float* __restrict__ out) {
  int b = blockIdx.x, t = threadIdx.x;  // 256
  out[b * 256 + t] = (t < DIM) ? h[b * DIM + t] : r[b * DIM + (t - DIM)];
}

// ---------------------------------------------------------------------------
extern "C" void kernel_launch(void* const* d_in, const int* in_sizes, int n_in,
                              void* d_out, int out_size, void* d_ws, size_t ws_size,
                              hipStream_t stream) {
  const float*     x      = (const float*)d_in[0];
  const long long* batch  = (const long long*)d_in[1];
  const float*     att_w  = (const float*)d_in[2];
  const float*     att_b  = (const float*)d_in[3];
  const float*     w_ih   = (const float*)d_in[4];
  const float*     w_hh   = (const float*)d_in[5];
  const float*     b_ih   = (const float*)d_in[6];
  const float*     b_hh   = (const float*)d_in[7];
  float* out = (float*)d_out;
  const int N = in_sizes[0] / DIM;

  // workspace carve-out (256B aligned slabs)
  size_t off = 0;
  auto carve = [&](size_t bytes) {
    void* p = (char*)d_ws + off;
    off = (off + bytes + 255) & ~(size_t)255;
    return p;
  };
  int*   seg_ptr   = (int*)  carve((NSEG + 1) * sizeof(int));
  float* q_star    = (float*)carve((size_t)NSEG * DIM * 4);
  float* h         = (float*)carve((size_t)NSEG * DIM * 4);
  float* c         = (float*)carve((size_t)NSEG * DIM * 4);
  float* v         = (float*)carve((size_t)NSEG * DIM * 4);
  float* r         = (float*)carve((size_t)NSEG * DIM * 4);
  float* lstm_in   = (float*)carve((size_t)NSEG * KPACK * 4);
  float* gates     = (float*)carve((size_t)NSEG * NGATE * 4);
  float* w_pack    = (float*)carve((size_t)NGATE * KPACK * 4);
  float* bias_pack = (float*)carve((size_t)NGATE * 4);
  float* scores    = (float*)carve((size_t)N * 4);
  (void)ws_size; (void)n_in; (void)out_size;

  build_seg_ptr<<<(N + 255) / 256, 256, 0, stream>>>(batch, seg_ptr, N);
  seg_mean<<<NSEG, DIM, 0, stream>>>(x, seg_ptr, q_star, h, c);
  pack_weights<<<NGATE, DIM, 0, stream>>>(w_ih, w_hh, b_ih, b_hh, w_pack, bias_pack);

  for (int step = 0; step < NSTEPS; ++step) {
    // v = q_star @ att_w  (M=2048, N=128, K=128): 1024 tiles -> 256 blocks x 4 waves
    wmma_gemm_f32<true><<<(NSEG / 16) * (DIM / 16) / 4, 128, 0, stream>>>(
        q_star, att_w, v, NSEG, DIM, DIM, DIM, DIM);
    attention_step<<<NSEG, DIM, 0, stream>>>(x, seg_ptr, v, att_b, q_star, scores, r);
    pack_lstm_in<<<NSEG, DIM, 0, stream>>>(q_star, r, h, lstm_in);
    // gates = lstm_in @ w_pack^T  (M=2048, N=512, K=384): 4096 tiles -> 1024 blocks
    wmma_gemm_f32<false><<<(NSEG / 16) * (NGATE / 16) / 4, 128, 0, stream>>>(
        lstm_in, w_pack, gates, NSEG, NGATE, KPACK, KPACK, KPACK);
    lstm_act<<<(NSEG * DIM) / 256, 256, 0, stream>>>(gates, bias_pack, h, c, q_star);
  }
  write_out<<<NSEG, 2 * DIM, 0, stream>>>(h, r, out);
}